// GCNReg_1211180778301
// MI455X (gfx1250) — compile-verified
//
#include <hip/hip_runtime.h>
#include <hip/hip_bf16.h>

typedef __attribute__((ext_vector_type(2))) float v2f;
typedef __attribute__((ext_vector_type(8))) float v8f;

#define NN_NODES 40000
#define NN_EDGES 640000
#define HIDN 128
#define NGRAPH 256

// ---------------------------------------------------------------------------
// Degree accumulation: one thread per edge, f32 atomics into deg arrays.
// ---------------------------------------------------------------------------
__global__ __launch_bounds__(256) void degree_kernel(const int* __restrict__ src,
                                                     const int* __restrict__ dst,
                                                     float* __restrict__ degO,
                                                     float* __restrict__ degI, int ne) {
    int e = blockIdx.x * 256 + threadIdx.x;
    if (e < ne) {
        atomicAdd(&degO[src[e]], 1.0f);
        atomicAdd(&degI[dst[e]], 1.0f);
    }
}

// norm = clip(deg,1)^-0.5, in place for both arrays
__global__ __launch_bounds__(256) void norm_kernel(float* __restrict__ degO,
                                                   float* __restrict__ degI, int n) {
    int i = blockIdx.x * 256 + threadIdx.x;
    if (i < n) {
        degO[i] = rsqrtf(fmaxf(degO[i], 1.0f));
        degI[i] = rsqrtf(fmaxf(degI[i], 1.0f));
    }
}

// ---------------------------------------------------------------------------
// WMMA GEMM: Y[M,128] = (X[M,128] * rowScale) @ W[128,128] (+bias)(ReLU)
// One block = 256 threads = 8 waves; block owns 16 rows, wave w owns cols
// 16w..16w+15. K-loop of 32 x V_WMMA_F32_16X16X4_F32 (exact f32 math).
// A-slab (16x128, row-scaled) staged in LDS once and shared by all 8 waves.
// ---------------------------------------------------------------------------
__global__ __launch_bounds__(256) void gemm128_wmma_kernel(
    const float* __restrict__ X, const float* __restrict__ rowScale,
    const float* __restrict__ W, const float* __restrict__ bias,
    float* __restrict__ Y, int M, int doRelu) {
    __shared__ float sX[16 * 128];

    const int rowBase = blockIdx.x * 16;
    const int tid = threadIdx.x;

    // Stage 16x128 A-slab with per-row scaling (2 float4 per thread)
    for (int i = tid; i < (16 * 128) / 4; i += 256) {
        int r = (i * 4) >> 7;
        int c = (i * 4) & 127;
        float4 v = *(const float4*)(X + (size_t)(rowBase + r) * 128 + c);
        float s = rowScale ? rowScale[rowBase + r] : 1.0f;
        v.x *= s; v.y *= s; v.z *= s; v.w *= s;
        *(float4*)(sX + i * 4) = v;
    }
    __syncthreads();

    const int wave = tid >> 5;          // 0..7
    const int lane = tid & 31;
    const int colBase = wave * 16;
    const int m = lane & 15;            // A row / B-C col within tile
    const int ks = lane >> 4;           // K-pair select (0 or 1)

    v8f acc = {};
#pragma unroll
    for (int k0 = 0; k0 < 128; k0 += 4) {
        const int kk = k0 + 2 * ks;
        v2f a;
        a.x = sX[m * 128 + kk];
        a.y = sX[m * 128 + kk + 1];
        v2f b;
        b.x = W[(size_t)kk * 128 + colBase + m];
        b.y = W[(size_t)(kk + 1) * 128 + colBase + m];
        // 8 args: (neg_a, A, neg_b, B, c_mod, C, reuse_a, reuse_b)
        acc = __builtin_amdgcn_wmma_f32_16x16x4_f32(false, a, false, b,
                                                    (short)0, acc, false, false);
    }

    // D layout: VGPR j -> M = j + 8*ks, N = colBase + (lane&15)
#pragma unroll
    for (int j = 0; j < 8; ++j) {
        int r = rowBase + j + 8 * ks;
        int col = colBase + m;
        float v = acc[j];
        if (bias) v += bias[col];
        if (doRelu) v = fmaxf(v, 0.0f);
        Y[(size_t)r * 128 + col] = v;
    }
}

// ---------------------------------------------------------------------------
// Edge aggregation: agg[dst[e], :] += hw[src[e], :].  One wave per edge,
// each lane moves a float4 (128 cols / 32 lanes).  All traffic is L2-resident
// (node matrix = 20.5 MB << 192 MB L2), atomics resolve at L2 atomic units.
// ---------------------------------------------------------------------------
__global__ __launch_bounds__(256) void aggregate_kernel(const int* __restrict__ src,
                                                        const int* __restrict__ dst,
                                                        const float* __restrict__ hw,
                                                        float* __restrict__ agg, int ne) {
    long long gid = (long long)blockIdx.x * 256 + threadIdx.x;
    int e = (int)(gid >> 5);
    if (e >= ne) return;
    int c = (int)(gid & 31) * 4;
    int s = src[e];
    int d = dst[e];
    float4 v = *(const float4*)(hw + (size_t)s * 128 + c);
    float* p = agg + (size_t)d * 128 + c;
    atomicAdd(p + 0, v.x);
    atomicAdd(p + 1, v.y);
    atomicAdd(p + 2, v.z);
    atomicAdd(p + 3, v.w);
}

// out = relu(agg * norm_d[row] + bias[col])
__global__ __launch_bounds__(256) void postagg_kernel(const float* __restrict__ agg,
                                                      const float* __restrict__ normd,
                                                      const float* __restrict__ bias,
                                                      float* __restrict__ out, int nn) {
    int gid = blockIdx.x * 256 + threadIdx.x;
    if (gid < nn * 128) {
        int r = gid >> 7;
        int c = gid & 127;
        out[gid] = fmaxf(agg[gid] * normd[r] + bias[c], 0.0f);
    }
}

// Per-graph sum + count via atomics (graph accumulator is 128 KB -> L2 hot)
__global__ __launch_bounds__(256) void pool_kernel(const float* __restrict__ h,
                                                   const int* __restrict__ gids,
                                                   float* __restrict__ sums,
                                                   float* __restrict__ counts, int nn) {
    long long gid = (long long)blockIdx.x * 256 + threadIdx.x;
    int node = (int)(gid >> 5);
    if (node >= nn) return;
    int part = (int)(gid & 31);
    int c = part * 4;
    int g = gids[node];
    float4 v = *(const float4*)(h + (size_t)node * 128 + c);
    float* p = sums + (size_t)g * 128 + c;
    atomicAdd(p + 0, v.x);
    atomicAdd(p + 1, v.y);
    atomicAdd(p + 2, v.z);
    atomicAdd(p + 3, v.w);
    if (part == 0) atomicAdd(&counts[g], 1.0f);
}

__global__ __launch_bounds__(256) void divide_kernel(float* __restrict__ sums,
                                                     const float* __restrict__ counts,
                                                     int total) {
    int gid = blockIdx.x * 256 + threadIdx.x;
    if (gid < total) sums[gid] /= fmaxf(counts[gid >> 7], 1.0f);
}

// out[g] = dot(m2[g,:], Wc3) + bc3[0].  One wave32 per graph, shuffle-reduce.
__global__ __launch_bounds__(32) void final_kernel(const float* __restrict__ m2,
                                                   const float* __restrict__ Wc3,
                                                   const float* __restrict__ bc3,
                                                   float* __restrict__ out) {
    int g = blockIdx.x;
    int lane = threadIdx.x;
    float s = 0.0f;
#pragma unroll
    for (int c = lane; c < 128; c += 32) s += m2[(size_t)g * 128 + c] * Wc3[c];
#pragma unroll
    for (int off = 16; off > 0; off >>= 1) s += __shfl_down(s, off, 32);
    if (lane == 0) out[g] = s + bc3[0];
}

// ---------------------------------------------------------------------------
extern "C" void kernel_launch(void* const* d_in, const int* in_sizes, int n_in,
                              void* d_out, int out_size, void* d_ws, size_t ws_size,
                              hipStream_t stream) {
    const float* h   = (const float*)d_in[0];
    const int*   src = (const int*)d_in[1];
    const int*   dst = (const int*)d_in[2];
    const int*   gid = (const int*)d_in[3];
    const float* W1  = (const float*)d_in[4];
    const float* b1  = (const float*)d_in[5];
    const float* W2  = (const float*)d_in[6];
    const float* b2  = (const float*)d_in[7];
    const float* Wc1 = (const float*)d_in[8];
    const float* bc1 = (const float*)d_in[9];
    const float* Wc2 = (const float*)d_in[10];
    const float* bc2 = (const float*)d_in[11];
    const float* Wc3 = (const float*)d_in[12];
    const float* bc3 = (const float*)d_in[13];
    float* out = (float*)d_out;

    const size_t NM = (size_t)NN_NODES * HIDN;
    float* w     = (float*)d_ws;
    float* bufA  = w;                       // hw1 -> h1 -> agg2
    float* bufB  = bufA + NM;               // agg1 -> hw2 -> h2
    float* degO  = bufB + NM;               // becomes norm_s in place
    float* degI  = degO + NN_NODES;         // becomes norm_d in place
    float* sums  = degI + NN_NODES;         // [256,128] graph sums -> hg
    float* cnts  = sums + (size_t)NGRAPH * HIDN;
    float* m1    = cnts + NGRAPH;
    float* m2    = m1 + (size_t)NGRAPH * HIDN;

    const int EB = (NN_EDGES + 255) / 256;                 // edge blocks (1/thread)
    const int AB = (int)(((long long)NN_EDGES * 32 + 255) / 256);  // aggregate blocks
    const int PB = (int)(((long long)NN_NODES * 32 + 255) / 256);  // pool blocks
    const int NEL = (NN_NODES * HIDN + 255) / 256;         // elementwise blocks
    const int GEMM_BLOCKS = NN_NODES / 16;                 // 2500

    // degrees -> norms
    hipMemsetAsync(degO, 0, 2 * NN_NODES * sizeof(float), stream);
    degree_kernel<<<EB, 256, 0, stream>>>(src, dst, degO, degI, NN_EDGES);
    norm_kernel<<<(NN_NODES + 255) / 256, 256, 0, stream>>>(degO, degI, NN_NODES);

    // ---- layer 1 ----
    gemm128_wmma_kernel<<<GEMM_BLOCKS, 256, 0, stream>>>(h, degO, W1, nullptr,
                                                         bufA, NN_NODES, 0);
    hipMemsetAsync(bufB, 0, NM * sizeof(float), stream);
    aggregate_kernel<<<AB, 256, 0, stream>>>(src, dst, bufA, bufB, NN_EDGES);
    postagg_kernel<<<NEL, 256, 0, stream>>>(bufB, degI, b1, bufA, NN_NODES);

    // ---- layer 2 ----
    gemm128_wmma_kernel<<<GEMM_BLOCKS, 256, 0, stream>>>(bufA, degO, W2, nullptr,
                                                         bufB, NN_NODES, 0);
    hipMemsetAsync(bufA, 0, NM * sizeof(float), stream);
    aggregate_kernel<<<AB, 256, 0, stream>>>(src, dst, bufB, bufA, NN_EDGES);
    postagg_kernel<<<NEL, 256, 0, stream>>>(bufA, degI, b2, bufB, NN_NODES);

    // ---- mean pooling ----
    hipMemsetAsync(sums, 0, (NGRAPH * HIDN + NGRAPH) * sizeof(float), stream);
    pool_kernel<<<PB, 256, 0, stream>>>(bufB, gid, sums, cnts, NN_NODES);
    divide_kernel<<<(NGRAPH * HIDN + 255) / 256, 256, 0, stream>>>(
        sums, cnts, NGRAPH * HIDN);

    // ---- MLP head (WMMA, bias + ReLU) ----
    gemm128_wmma_kernel<<<NGRAPH / 16, 256, 0, stream>>>(sums, nullptr, Wc1, bc1,
                                                         m1, NGRAPH, 1);
    gemm128_wmma_kernel<<<NGRAPH / 16, 256, 0, stream>>>(m1, nullptr, Wc2, bc2,
                                                         m2, NGRAPH, 1);
    final_kernel<<<NGRAPH, 32, 0, stream>>>(m2, Wc3, bc3, out);
}